// BinNormInf_86775519248463
// MI455X (gfx1250) — compile-verified
//
#include <hip/hip_runtime.h>
#include <hip/hip_bf16.h>

typedef unsigned int u32;
typedef unsigned int  u32x4 __attribute__((ext_vector_type(4)));
typedef int           i32x4 __attribute__((ext_vector_type(4)));
typedef int           i32x8 __attribute__((ext_vector_type(8)));
typedef float         f32x4 __attribute__((ext_vector_type(4)));

#define WAVES_PER_BLOCK 8
#define TILE_PIX   4
#define CHANNELS   256
#define TILE_WORDS (TILE_PIX * CHANNELS)   // 1024 dwords = 4KB
#define TILE_BYTES (TILE_WORDS * 4)

// float bits -> order-preserving unsigned key
__device__ __forceinline__ u32 f2key(u32 u) {
    u32 m = ((u32)((int)u >> 31)) | 0x80000000u;
    return u ^ m;
}

template <int N>
__device__ __forceinline__ void tensor_wait() {
#if __has_builtin(__builtin_amdgcn_s_wait_tensorcnt)
    __builtin_amdgcn_s_wait_tensorcnt(N);
#endif
}

// Issue one TDM load of TILE_BYTES (contiguous) from global to LDS.
// lds_byte_addr: LDS byte offset of destination; ldsdst: generic pointer (fallback path).
__device__ __forceinline__ void tdm_load_tile(const float* gptr, u32 lds_byte_addr,
                                              u32* ldsdst, u32 lane) {
#if __has_builtin(__builtin_amdgcn_tensor_load_to_lds)
    unsigned long long ga = (unsigned long long)gptr;
    u32 ga_lo = __builtin_amdgcn_readfirstlane((u32)ga);
    u32 ga_hi = __builtin_amdgcn_readfirstlane((u32)(ga >> 32));
    u32 la    = __builtin_amdgcn_readfirstlane(lds_byte_addr);

    // D# group 0: count=1 | lds_addr | global_addr[56:0] | type=2
    u32x4 g0;
    g0.x = 1u;                                   // count=1, user descriptor
    g0.y = la;                                   // lds_addr (bytes)
    g0.z = ga_lo;                                // global_addr[31:0]
    g0.w = (ga_hi & 0x01FFFFFFu) | (2u << 30);   // global_addr[56:32] | type=2

    // D# group 1: data_size=4B, 1-row tile of TILE_WORDS elements
    i32x8 g1;
    g1[0] = (int)(2u << 16);                                   // wg_mask=0, data_size=4B
    g1[1] = (int)(((u32)TILE_WORDS & 0xFFFFu) << 16);          // tensor_dim0[15:0]
    g1[2] = (int)(((u32)TILE_WORDS >> 16) | (1u << 16));       // tensor_dim0[31:16] | tensor_dim1=1 lo
    g1[3] = (int)((u32)TILE_WORDS << 16);                      // tensor_dim1 hi=0 | tile_dim0
    g1[4] = 1;                                                 // tile_dim1=1, tile_dim2=0
    g1[5] = TILE_WORDS;                                        // tensor_dim0_stride lo32
    g1[6] = 0;                                                 // stride0 hi | stride1 lo
    g1[7] = 0;                                                 // stride1 hi

    i32x4 g2 = {0, 0, 0, 0};
    i32x4 g3 = {0, 0, 0, 0};
    i32x8 g4 = {0, 0, 0, 0, 0, 0, 0, 0};
    __builtin_amdgcn_tensor_load_to_lds(g0, g1, g2, g3, g4, 0);
#else
    // Fallback: synchronous lane-parallel copy (8 x b128 per lane)
    const u32x4* g = (const u32x4*)gptr;
    u32x4* l = (u32x4*)ldsdst;
    #pragma unroll
    for (int i = 0; i < 8; ++i) {
        l[i * 32 + lane] = g[i * 32 + lane];
    }
#endif
}

__global__ __launch_bounds__(256) void topk_mask_kernel(
    const float* __restrict__ x, const int* __restrict__ kp,
    float* __restrict__ out, int npix) {
    __shared__ u32 lds[2 * WAVES_PER_BLOCK * TILE_WORDS];   // 64 KB

    const int kk   = kp[0];
    const u32 lane = threadIdx.x & 31u;
    const u32 wave = __builtin_amdgcn_readfirstlane(threadIdx.x >> 5);

    const int nwaves = (int)(gridDim.x * WAVES_PER_BLOCK);
    const int wid    = (int)(blockIdx.x * WAVES_PER_BLOCK + wave);
    const int ntiles = npix / TILE_PIX;

    const u32 lds_base = (u32)(unsigned long long)&lds[0];   // low 32 bits = LDS offset
    const u32 woff0 = (wave * 2u + 0u) * TILE_WORDS;
    const u32 woff1 = (wave * 2u + 1u) * TILE_WORDS;

    int cur = wid;
    int parity = 0;
    if (cur < ntiles) {
        tdm_load_tile(x + (size_t)cur * TILE_WORDS, lds_base + woff0 * 4u,
                      &lds[woff0], lane);
    }

    for (; cur < ntiles; cur += nwaves) {
        const int nxt = cur + nwaves;
        const u32 woff_cur = parity ? woff1 : woff0;
        const u32 woff_nxt = parity ? woff0 : woff1;

        asm volatile("" ::: "memory");
        if (nxt < ntiles) {
            tdm_load_tile(x + (size_t)nxt * TILE_WORDS, lds_base + woff_nxt * 4u,
                          &lds[woff_nxt], lane);
            tensor_wait<1>();   // tile `cur` finished, tile `nxt` may be in flight
        } else {
            tensor_wait<0>();
        }
        asm volatile("" ::: "memory");

        #pragma unroll
        for (int p = 0; p < TILE_PIX; ++p) {
            const int pix = cur * TILE_PIX + p;
            const u32* pl = &lds[woff_cur + (u32)p * CHANNELS + lane * 8u];
            u32x4 a = *(const u32x4*)(pl);        // ds_load_b128
            u32x4 b = *(const u32x4*)(pl + 4);    // ds_load_b128

            u32 key[8];
            key[0] = f2key(a.x); key[1] = f2key(a.y);
            key[2] = f2key(a.z); key[3] = f2key(a.w);
            key[4] = f2key(b.x); key[5] = f2key(b.y);
            key[6] = f2key(b.z); key[7] = f2key(b.w);

            // Bit-serial radix select of the kk-th largest key (wave-uniform scalar t).
            u32 t = 0u;
            for (int bit = 31; bit >= 0; --bit) {
                const u32 cand = t | (1u << bit);
                unsigned long long m0 =
                    ((unsigned long long)__builtin_amdgcn_ballot_w32(key[1] >= cand) << 32) |
                     (unsigned long long)__builtin_amdgcn_ballot_w32(key[0] >= cand);
                unsigned long long m1 =
                    ((unsigned long long)__builtin_amdgcn_ballot_w32(key[3] >= cand) << 32) |
                     (unsigned long long)__builtin_amdgcn_ballot_w32(key[2] >= cand);
                unsigned long long m2 =
                    ((unsigned long long)__builtin_amdgcn_ballot_w32(key[5] >= cand) << 32) |
                     (unsigned long long)__builtin_amdgcn_ballot_w32(key[4] >= cand);
                unsigned long long m3 =
                    ((unsigned long long)__builtin_amdgcn_ballot_w32(key[7] >= cand) << 32) |
                     (unsigned long long)__builtin_amdgcn_ballot_w32(key[6] >= cand);
                const u32 c = (u32)(__builtin_popcountll(m0) + __builtin_popcountll(m1) +
                                    __builtin_popcountll(m2) + __builtin_popcountll(m3));
                t = (c >= (u32)kk) ? cand : t;
            }

            // Count strictly-greater; remaining slots go to equal keys in index order.
            unsigned long long q0 =
                ((unsigned long long)__builtin_amdgcn_ballot_w32(key[1] > t) << 32) |
                 (unsigned long long)__builtin_amdgcn_ballot_w32(key[0] > t);
            unsigned long long q1 =
                ((unsigned long long)__builtin_amdgcn_ballot_w32(key[3] > t) << 32) |
                 (unsigned long long)__builtin_amdgcn_ballot_w32(key[2] > t);
            unsigned long long q2 =
                ((unsigned long long)__builtin_amdgcn_ballot_w32(key[5] > t) << 32) |
                 (unsigned long long)__builtin_amdgcn_ballot_w32(key[4] > t);
            unsigned long long q3 =
                ((unsigned long long)__builtin_amdgcn_ballot_w32(key[7] > t) << 32) |
                 (unsigned long long)__builtin_amdgcn_ballot_w32(key[6] > t);
            const int cgt = __builtin_popcountll(q0) + __builtin_popcountll(q1) +
                            __builtin_popcountll(q2) + __builtin_popcountll(q3);
            const int need = kk - cgt;   // #equal keys to admit (stable by index)

            // Cross-lane count of equal keys in strictly lower lanes (all 8 slots).
            u32 e[8];
            #pragma unroll
            for (int s = 0; s < 8; ++s)
                e[s] = __builtin_amdgcn_ballot_w32(key[s] == t);
            u32 base = 0u;
            #pragma unroll
            for (int s = 0; s < 8; ++s)
                base = __builtin_amdgcn_mbcnt_lo(e[s], base);

            float r[8];
            u32 pre = base;   // equal-keys with smaller channel index (lane-major order)
            #pragma unroll
            for (int s = 0; s < 8; ++s) {
                const bool gt = key[s] > t;
                const bool eq = key[s] == t;
                r[s] = (gt || (eq && ((int)pre < need))) ? 1.0f : 0.0f;
                pre += eq ? 1u : 0u;
            }

            if (pix < npix) {
                float* op = out + (size_t)pix * CHANNELS + lane * 8u;
                f32x4 v0 = { r[0], r[1], r[2], r[3] };
                f32x4 v1 = { r[4], r[5], r[6], r[7] };
                __builtin_nontemporal_store(v0, (f32x4*)op);
                __builtin_nontemporal_store(v1, (f32x4*)(op + 4));
            }
        }
        parity ^= 1;
    }
}

extern "C" void kernel_launch(void* const* d_in, const int* in_sizes, int n_in,
                              void* d_out, int out_size, void* d_ws, size_t ws_size,
                              hipStream_t stream) {
    const float* x  = (const float*)d_in[0];
    const int*   kp = (const int*)d_in[1];
    float*       out = (float*)d_out;
    const int npix = in_sizes[0] / CHANNELS;   // 307200 for 1x480x640x256

    dim3 grid(960);
    dim3 block(256);
    topk_mask_kernel<<<grid, block, 0, stream>>>(x, kp, out, npix);
}